// SpatialTransformer_19688130085661
// MI455X (gfx1250) — compile-verified
//
#include <hip/hip_runtime.h>

typedef float v2f __attribute__((ext_vector_type(2)));
typedef float v8f __attribute__((ext_vector_type(8)));

#define NN 100000
#define EE 600000
#define CC 128
#define HH 4
#define DDIM 32

// ---------- helpers ----------
__device__ __forceinline__ float wsum(float v) {
  v += __shfl_xor(v, 16, 32);
  v += __shfl_xor(v, 8, 32);
  v += __shfl_xor(v, 4, 32);
  v += __shfl_xor(v, 2, 32);
  v += __shfl_xor(v, 1, 32);
  return v;
}
// order-preserving float->uint key for atomic max over signed floats
__device__ __forceinline__ unsigned fkey(float x) {
  unsigned u = __float_as_uint(x);
  return (u & 0x80000000u) ? ~u : (u | 0x80000000u);
}
__device__ __forceinline__ float funkey(unsigned k) {
  unsigned u = (k & 0x80000000u) ? (k & 0x7fffffffu) : ~k;
  return __uint_as_float(u);
}
__device__ __forceinline__ void fadd_atomic(float* p, float v) {
  __hip_atomic_fetch_add(p, v, __ATOMIC_RELAXED, __HIP_MEMORY_SCOPE_AGENT);
}

// ---------- Kernel A: fused node GEMMs (Q,K,V,skip) via WMMA f32 16x16x4 ----------
__global__ __launch_bounds__(256)
void node_gemm_kernel(const float* __restrict__ x,
                      const float* __restrict__ Wq, const float* __restrict__ bq,
                      const float* __restrict__ Wk, const float* __restrict__ bk,
                      const float* __restrict__ Wv, const float* __restrict__ bv,
                      const float* __restrict__ Wsk, const float* __restrict__ bsk,
                      float* __restrict__ qb, float* __restrict__ kb,
                      float* __restrict__ vb, float* __restrict__ xrb) {
  __shared__ float xs[16 * 132];   // 16 rows x 128 cols, padded to avoid bank conflicts
  const int rt  = blockIdx.x;      // 16-row tile
  const int tid = threadIdx.x;
  for (int i = tid; i < 16 * 128; i += 256) {
    int r = i >> 7, c = i & 127;
    xs[r * 132 + c] = x[(size_t)(rt * 16 + r) * 128 + c];
  }
  __syncthreads();

  const int wv   = tid >> 5;       // wave id: owns output cols [16wv,16wv+16)
  const int lane = tid & 31;
  const int hi   = lane >> 4;      // 0: K=0,1  1: K=2,3
  const int l16  = lane & 15;
  const int n    = wv * 16 + l16;  // output column
  const float* Wm[4]   = {Wq, Wk, Wv, Wsk};
  const float* Bm[4]   = {bq, bk, bv, bsk};
  float*       Om[4]   = {qb, kb, vb, xrb};

  #pragma unroll
  for (int m = 0; m < 4; ++m) {
    const float* W = Wm[m];
    v8f acc = {0.f, 0.f, 0.f, 0.f, 0.f, 0.f, 0.f, 0.f};
    #pragma unroll
    for (int k0 = 0; k0 < 128; k0 += 4) {
      int kk = k0 + hi * 2;
      v2f a, b;
      a.x = xs[l16 * 132 + kk];
      a.y = xs[l16 * 132 + kk + 1];
      b.x = W[(size_t)n * 128 + kk];      // B[k][n] = W[n][k]
      b.y = W[(size_t)n * 128 + kk + 1];
      acc = __builtin_amdgcn_wmma_f32_16x16x4_f32(false, a, false, b,
                                                  (short)0, acc, false, false);
    }
    float bias = Bm[m][n];
    float* op  = Om[m];
    #pragma unroll
    for (int j = 0; j < 8; ++j) {
      int row = rt * 16 + ((lane < 16) ? j : (j + 8));
      op[(size_t)row * 128 + n] = acc[j] + bias;
    }
  }
}

// ---------- edge helper: recompute ea (Linear(1,32)+LN+ReLU) and e = ea @ We^T ----------
// one wave per edge; WeT staged in LDS as [j][c] (32 x 128)
__device__ __forceinline__ void edge_feat(const float* __restrict__ WeT,
                                          float* __restrict__ eaS, int wedge, int lane,
                                          float ewv,
                                          const float* __restrict__ Wep,
                                          const float* __restrict__ bep,
                                          const float* __restrict__ gep,
                                          const float* __restrict__ bln,
                                          float& e0, float& e1, float& e2, float& e3) {
  float t = Wep[lane] * ewv + bep[lane];
  float mean = wsum(t) * (1.f / 32.f);
  float d = t - mean;
  float var = wsum(d * d) * (1.f / 32.f);
  float ea = d * rsqrtf(var + 1e-5f) * gep[lane] + bln[lane];
  ea = fmaxf(ea, 0.f);
  eaS[wedge * 32 + lane] = ea;   // wave-private slot; same-wave DS ops stay in order
  e0 = e1 = e2 = e3 = 0.f;
  #pragma unroll 8
  for (int j = 0; j < 32; ++j) {
    float eaj = eaS[wedge * 32 + j];
    e0 += WeT[j * 128 + lane]      * eaj;
    e1 += WeT[j * 128 + lane + 32] * eaj;
    e2 += WeT[j * 128 + lane + 64] * eaj;
    e3 += WeT[j * 128 + lane + 96] * eaj;
  }
}

// ---------- Kernel B: per-edge attention logits + segment max ----------
__global__ __launch_bounds__(256)
void edge_pass1_kernel(const int* __restrict__ ei, const float* __restrict__ ew,
                       const float* __restrict__ Wep, const float* __restrict__ bep,
                       const float* __restrict__ gep, const float* __restrict__ bln,
                       const float* __restrict__ We,
                       const float* __restrict__ qb, const float* __restrict__ kb,
                       float* __restrict__ alpha, unsigned* __restrict__ amaxKey) {
  __shared__ float WeT[32 * 128];
  __shared__ float eaS[8 * 32];
  const int tid = threadIdx.x;
  for (int i = tid; i < 4096; i += 256) {    // We[c][j] -> WeT[j][c]
    int c = i >> 5, j = i & 31;
    WeT[j * 128 + c] = We[i];
  }
  __syncthreads();
  const int wedge = tid >> 5, lane = tid & 31;
  const int e = blockIdx.x * 8 + wedge;
  const int src = ei[e];
  const int dst = ei[EE + e];
  float e0, e1, e2, e3;
  edge_feat(WeT, eaS, wedge, lane, ew[e], Wep, bep, gep, bln, e0, e1, e2, e3);

  const float* qr = qb + (size_t)dst * 128;
  const float* kr = kb + (size_t)src * 128;
  float p0 = qr[lane]      * (kr[lane]      + e0);
  float p1 = qr[lane + 32] * (kr[lane + 32] + e1);
  float p2 = qr[lane + 64] * (kr[lane + 64] + e2);
  float p3 = qr[lane + 96] * (kr[lane + 96] + e3);
  const float s = 0.17677669529663687f;  // 1/sqrt(32)
  float a0 = wsum(p0) * s, a1 = wsum(p1) * s, a2 = wsum(p2) * s, a3 = wsum(p3) * s;
  if (lane == 0) {
    float4 av = {a0, a1, a2, a3};
    *(float4*)(alpha + (size_t)e * 4) = av;
    atomicMax(amaxKey + (size_t)dst * 4 + 0, fkey(a0));
    atomicMax(amaxKey + (size_t)dst * 4 + 1, fkey(a1));
    atomicMax(amaxKey + (size_t)dst * 4 + 2, fkey(a2));
    atomicMax(amaxKey + (size_t)dst * 4 + 3, fkey(a3));
  }
}

// ---------- Kernel C: exp, denom, weighted-value scatter ----------
__global__ __launch_bounds__(256)
void edge_pass2_kernel(const int* __restrict__ ei, const float* __restrict__ ew,
                       const float* __restrict__ Wep, const float* __restrict__ bep,
                       const float* __restrict__ gep, const float* __restrict__ bln,
                       const float* __restrict__ We,
                       const float* __restrict__ vb,
                       const float* __restrict__ alpha, const unsigned* __restrict__ amaxKey,
                       float* __restrict__ denom, float* __restrict__ outnum) {
  __shared__ float WeT[32 * 128];
  __shared__ float eaS[8 * 32];
  const int tid = threadIdx.x;
  for (int i = tid; i < 4096; i += 256) {
    int c = i >> 5, j = i & 31;
    WeT[j * 128 + c] = We[i];
  }
  __syncthreads();
  const int wedge = tid >> 5, lane = tid & 31;
  const int e = blockIdx.x * 8 + wedge;
  const int src = ei[e];
  const int dst = ei[EE + e];
  float e0, e1, e2, e3;
  edge_feat(WeT, eaS, wedge, lane, ew[e], Wep, bep, gep, bln, e0, e1, e2, e3);

  const float4 av = *(const float4*)(alpha + (size_t)e * 4);
  float m0 = funkey(amaxKey[(size_t)dst * 4 + 0]);
  float m1 = funkey(amaxKey[(size_t)dst * 4 + 1]);
  float m2 = funkey(amaxKey[(size_t)dst * 4 + 2]);
  float m3 = funkey(amaxKey[(size_t)dst * 4 + 3]);
  float x0 = __expf(av.x - m0), x1 = __expf(av.y - m1);
  float x2 = __expf(av.z - m2), x3 = __expf(av.w - m3);
  if (lane == 0) {
    fadd_atomic(denom + (size_t)dst * 4 + 0, x0);
    fadd_atomic(denom + (size_t)dst * 4 + 1, x1);
    fadd_atomic(denom + (size_t)dst * 4 + 2, x2);
    fadd_atomic(denom + (size_t)dst * 4 + 3, x3);
  }
  const float* vr = vb + (size_t)src * 128;
  float* orow = outnum + (size_t)dst * 128;
  fadd_atomic(orow + lane,      x0 * (vr[lane]      + e0));
  fadd_atomic(orow + lane + 32, x1 * (vr[lane + 32] + e1));
  fadd_atomic(orow + lane + 64, x2 * (vr[lane + 64] + e2));
  fadd_atomic(orow + lane + 96, x3 * (vr[lane + 96] + e3));
}

// ---------- Kernel D: normalize, beta-gated skip, output LayerNorm ----------
__global__ __launch_bounds__(256)
void node_final_kernel(float* __restrict__ out,            // holds numerator, overwritten
                       const float* __restrict__ denom,
                       const float* __restrict__ xrb,
                       const float* __restrict__ Wb,
                       const float* __restrict__ gout, const float* __restrict__ bout) {
  const int tid = threadIdx.x;
  const int wv = tid >> 5, lane = tid & 31;
  const int n = blockIdx.x * 8 + wv;
  float o[4], xr[4];
  float z = 0.f;
  #pragma unroll
  for (int i = 0; i < 4; ++i) {
    int c = i * 32 + lane;
    float num = out[(size_t)n * 128 + c];
    float den = denom[(size_t)n * 4 + i] + 1e-16f;
    o[i]  = num / den;
    xr[i] = xrb[(size_t)n * 128 + c];
    float wA = Wb[c] + Wb[256 + c];        // multiplies out
    float wB = Wb[128 + c] - Wb[256 + c];  // multiplies x_r
    z += o[i] * wA + xr[i] * wB;
  }
  z = wsum(z);
  float beta = 1.f / (1.f + __expf(-z));
  float u[4], s = 0.f;
  #pragma unroll
  for (int i = 0; i < 4; ++i) { u[i] = beta * xr[i] + (1.f - beta) * o[i]; s += u[i]; }
  float mean = wsum(s) * (1.f / 128.f);
  float vv = 0.f;
  #pragma unroll
  for (int i = 0; i < 4; ++i) { float d = u[i] - mean; vv += d * d; }
  vv = wsum(vv) * (1.f / 128.f);
  float inv = rsqrtf(vv + 1e-5f);
  #pragma unroll
  for (int i = 0; i < 4; ++i) {
    int c = i * 32 + lane;
    out[(size_t)n * 128 + c] = (u[i] - mean) * inv * gout[c] + bout[c];
  }
}

// ---------- launch ----------
extern "C" void kernel_launch(void* const* d_in, const int* in_sizes, int n_in,
                              void* d_out, int out_size, void* d_ws, size_t ws_size,
                              hipStream_t stream) {
  const float* x    = (const float*)d_in[0];
  const int*   ei   = (const int*)  d_in[1];
  const float* ew   = (const float*)d_in[2];
  const float* Wep  = (const float*)d_in[3];
  const float* bep  = (const float*)d_in[4];
  const float* gep  = (const float*)d_in[5];
  const float* bln  = (const float*)d_in[6];
  const float* Wq   = (const float*)d_in[7];
  const float* bq   = (const float*)d_in[8];
  const float* Wk   = (const float*)d_in[9];
  const float* bk   = (const float*)d_in[10];
  const float* Wv   = (const float*)d_in[11];
  const float* bv   = (const float*)d_in[12];
  const float* We   = (const float*)d_in[13];
  const float* Wsk  = (const float*)d_in[14];
  const float* bsk  = (const float*)d_in[15];
  const float* Wb   = (const float*)d_in[16];
  const float* gout = (const float*)d_in[17];
  const float* bout = (const float*)d_in[18];
  float* out = (float*)d_out;

  char* ws = (char*)d_ws;
  size_t off = 0;
  float* qb  = (float*)(ws + off); off += (size_t)NN * 128 * 4;
  float* kb  = (float*)(ws + off); off += (size_t)NN * 128 * 4;
  float* vb  = (float*)(ws + off); off += (size_t)NN * 128 * 4;
  float* xrb = (float*)(ws + off); off += (size_t)NN * 128 * 4;
  float*    alpha   = (float*)(ws + off);    off += (size_t)EE * 4 * 4;
  unsigned* amaxKey = (unsigned*)(ws + off); off += (size_t)NN * 4 * 4;
  float*    denom   = (float*)(ws + off);    off += (size_t)NN * 4 * 4;

  hipMemsetAsync(amaxKey, 0, (size_t)NN * 4 * 4, stream);
  hipMemsetAsync(denom,   0, (size_t)NN * 4 * 4, stream);
  hipMemsetAsync(d_out,   0, (size_t)NN * 128 * 4, stream);

  node_gemm_kernel<<<NN / 16, 256, 0, stream>>>(x, Wq, bq, Wk, bk, Wv, bv, Wsk, bsk,
                                                qb, kb, vb, xrb);
  edge_pass1_kernel<<<EE / 8, 256, 0, stream>>>(ei, ew, Wep, bep, gep, bln, We,
                                                qb, kb, alpha, amaxKey);
  edge_pass2_kernel<<<EE / 8, 256, 0, stream>>>(ei, ew, Wep, bep, gep, bln, We,
                                                vb, alpha, amaxKey, denom, out);
  node_final_kernel<<<NN / 8, 256, 0, stream>>>(out, denom, xrb, Wb, gout, bout);
}